// MLEFeatureExtractor_71683004170598
// MI455X (gfx1250) — compile-verified
//
#include <hip/hip_runtime.h>
#include <hip/hip_bf16.h>
#include <math.h>

typedef __attribute__((ext_vector_type(2))) float v2f;
typedef __attribute__((ext_vector_type(4))) float v4f;
typedef __attribute__((ext_vector_type(8))) float v8f;

#define B_DIM 128
#define S_DIM 4096
#define DK_IN 128
#define DO_UT 64
#define DELAYN 2
#define EPSF 1e-6f

// ---------------------------------------------------------------------------
// Stage 1 (HBM-bound): t[b,s] = w2 . relu(x[b,s,:] @ w1^T + b1) + b2
// A = w1 zero-padded to 16 rows (from LDS), B = 16 x-rows (N dim, from HBM).
// Result rows M=0,1 land in c[0], c[1] at lane = position -> no shuffles.
// K is repartitioned so each lane streams one v4f per iteration feeding
// two v_wmma_f32_16x16x4_f32 (A and B use the identical K mapping).
// ---------------------------------------------------------------------------
__global__ __launch_bounds__(256) void stage1_wmma(
    const float* __restrict__ x, const float* __restrict__ w1,
    const float* __restrict__ b1, const float* __restrict__ w2,
    const float* __restrict__ b2, float* __restrict__ t_out)
{
  __shared__ float lds_w1[16 * DK_IN];          // 8 KB: rows 0,1 = w1; rest 0
  for (int i = threadIdx.x; i < 16 * DK_IN; i += 256)
    lds_w1[i] = (i < 2 * DK_IN) ? w1[i] : 0.0f;
  __syncthreads();

  const int  lane      = threadIdx.x & 31;
  const int  waveInBlk = threadIdx.x >> 5;
  const long posBase   = ((long)blockIdx.x * 8 + waveInBlk) * 16;

  const int khalf = lane >> 4;                  // K sub-block within each group of 8
  const int mrow  = lane & 15;                  // A: M (w1 row) / B: N (position)
  const v4f* xrow = (const v4f*)(x + (posBase + mrow) * DK_IN + 4 * khalf);
  const v4f* arow = (const v4f*)(lds_w1 + mrow * DK_IN + 4 * khalf);

  v8f c = {};
  #pragma unroll 2
  for (int i = 0; i < 16; ++i) {
    v4f bv = __builtin_nontemporal_load(&xrow[2 * i]);  // 16B/lane x stream (TH=NT)
    v4f av = arow[2 * i];                               // ds_load_b128
    v2f a0 = {av.x, av.y}, b0 = {bv.x, bv.y};
    v2f a1 = {av.z, av.w}, b1v = {bv.z, bv.w};
    c = __builtin_amdgcn_wmma_f32_16x16x4_f32(
            false, a0, false, b0, (short)0, c, false, false);
    c = __builtin_amdgcn_wmma_f32_16x16x4_f32(
            false, a1, false, b1v, (short)0, c, false, false);
  }

  // D layout: VGPR r holds M=r for lanes 0-15 (N=lane). We need M=0 and M=1.
  if (lane < 16) {
    float h0 = fmaxf(c[0] + b1[0], 0.0f);
    float h1 = fmaxf(c[1] + b1[1], 0.0f);
    t_out[posBase + lane] = fmaf(w2[1], h1, fmaf(w2[0], h0, b2[0]));
  }
}

// ---------------------------------------------------------------------------
// Stage 2: per-batch mean / std(ddof=1) of log(|t[s+2]-t[s]| + eps)
// ---------------------------------------------------------------------------
__global__ __launch_bounds__(256) void stage2_reduce(
    const float* __restrict__ t, float* __restrict__ feat)
{
  __shared__ float s_sum[256];
  __shared__ float s_sq[256];
  const int  b  = blockIdx.x;
  const int  n  = S_DIM - DELAYN;               // 4094
  const float* tb = t + (long)b * S_DIM;

  float sum = 0.0f, sq = 0.0f;
  for (int s = threadIdx.x; s < n; s += 256) {
    float d  = fabsf(tb[s + DELAYN] - tb[s]);   // E==1 -> L2 norm is |.|
    float ld = logf(d + EPSF);
    sum += ld;
    sq  += ld * ld;
  }
  s_sum[threadIdx.x] = sum;
  s_sq[threadIdx.x]  = sq;
  __syncthreads();
  #pragma unroll
  for (int off = 128; off > 0; off >>= 1) {
    if (threadIdx.x < off) {
      s_sum[threadIdx.x] += s_sum[threadIdx.x + off];
      s_sq[threadIdx.x]  += s_sq[threadIdx.x + off];
    }
    __syncthreads();
  }
  if (threadIdx.x == 0) {
    float mean = s_sum[0] / (float)n;
    float var  = (s_sq[0] - s_sum[0] * mean) / (float)(n - 1);
    feat[b * 2 + 0] = mean;
    feat[b * 2 + 1] = sqrtf(fmaxf(var, 0.0f));
  }
}

// ---------------------------------------------------------------------------
// Stage 3: out[b, :, :] = tanh(feat[b] @ w3^T + b3) broadcast over rows.
// One wave per 16x16 tile; K padded from 2 to 4 with zeros.
// ---------------------------------------------------------------------------
__global__ __launch_bounds__(32) void stage3_wmma(
    const float* __restrict__ feat, const float* __restrict__ w3,
    const float* __restrict__ b3, float* __restrict__ out)
{
  const int lane = threadIdx.x & 31;
  const int mt   = blockIdx.x;                  // batch tile   0..7
  const int nt   = blockIdx.y;                  // output tile  0..3
  const int idx  = lane & 15;

  v2f a = {}, b = {};
  if (lane < 16) {                              // lanes 16-31: K=2,3 -> zero pad
    a.x = feat[(mt * 16 + idx) * 2 + 0];        // A[M][K=0]
    a.y = feat[(mt * 16 + idx) * 2 + 1];        // A[M][K=1]
    b.x = w3[(nt * 16 + idx) * 2 + 0];          // B[K=0][N] = w3[N][0]
    b.y = w3[(nt * 16 + idx) * 2 + 1];          // B[K=1][N] = w3[N][1]
  }
  v8f c = {};
  c = __builtin_amdgcn_wmma_f32_16x16x4_f32(
          false, a, false, b, (short)0, c, false, false);

  const int   ncol = nt * 16 + idx;
  const float bias = b3[ncol];
  #pragma unroll
  for (int r = 0; r < 8; ++r) {
    int   m   = mt * 16 + r + ((lane >> 4) << 3);
    float val = tanhf(c[r] + bias);
    float* dst = out + (long)m * (DO_UT * DO_UT) + ncol;
    for (int j = 0; j < DO_UT; ++j)             // identical broadcast rows
      dst[(long)j * DO_UT] = val;
  }
}

extern "C" void kernel_launch(void* const* d_in, const int* in_sizes, int n_in,
                              void* d_out, int out_size, void* d_ws, size_t ws_size,
                              hipStream_t stream) {
  const float* x  = (const float*)d_in[0];
  const float* w1 = (const float*)d_in[1];
  const float* b1 = (const float*)d_in[2];
  const float* w2 = (const float*)d_in[3];
  const float* b2 = (const float*)d_in[4];
  const float* w3 = (const float*)d_in[5];
  const float* b3 = (const float*)d_in[6];
  float* out = (float*)d_out;

  // Workspace: t (B*S floats = 2 MB) then feat (B*2 floats).
  float* t_ws    = (float*)d_ws;
  float* feat_ws = t_ws + (size_t)B_DIM * S_DIM;

  // 524288 rows / (16 rows/wave * 8 waves/block) = 4096 blocks, exact tiling.
  stage1_wmma<<<(B_DIM * S_DIM) / (16 * 8), 256, 0, stream>>>(x, w1, b1, w2, b2, t_ws);
  stage2_reduce<<<B_DIM, 256, 0, stream>>>(t_ws, feat_ws);
  dim3 g3(B_DIM / 16, DO_UT / 16);
  stage3_wmma<<<g3, 32, 0, stream>>>(feat_ws, w3, b3, out);
}